// GCN_53626961658271
// MI455X (gfx1250) — compile-verified
//
#include <hip/hip_runtime.h>

typedef __attribute__((ext_vector_type(2))) float v2f;
typedef __attribute__((ext_vector_type(4))) float v4f;
typedef __attribute__((ext_vector_type(8))) float v8f;

#define N_NODES 50000
#define N_EDGES 800000
#define F_INF   128
#define HIDDENF 256
#define F_OUTF  64

// ---------------- utility kernels ----------------

__global__ void k_zero(float* __restrict__ p, int n) {
    int i = blockIdx.x * blockDim.x + threadIdx.x;
    if (i < n) p[i] = 0.0f;
}

__global__ void k_deg(const long long* __restrict__ col, float* __restrict__ deg, int E) {
    int e = blockIdx.x * blockDim.x + threadIdx.x;
    if (e < E) atomicAdd(&deg[col[e]], 1.0f);
}

__global__ void k_dinv(const float* __restrict__ deg, float* __restrict__ dinv, int n) {
    int i = blockIdx.x * blockDim.x + threadIdx.x;
    if (i < n) {
        float d = deg[i];
        dinv[i] = (d > 0.0f) ? rsqrtf(d) : 0.0f;
    }
}

__global__ void k_norm(const long long* __restrict__ row, const long long* __restrict__ col,
                       const float* __restrict__ dinv, float* __restrict__ norm, int E) {
    int e = blockIdx.x * blockDim.x + threadIdx.x;
    if (e < E) norm[e] = dinv[row[e]] * dinv[col[e]];
}

// out[i*F_OUTF + j] = b2[j]
__global__ void k_init_out(float* __restrict__ out, const float* __restrict__ b2, int total) {
    int i = blockIdx.x * blockDim.x + threadIdx.x;
    if (i < total) out[i] = b2[i & (F_OUTF - 1)];
}

// dst[col[e]] += src[row[e]] * norm[e], F = 1<<FSHIFT features, 4 per thread
template <int FSHIFT>
__global__ void k_aggregate(const float* __restrict__ src,
                            const long long* __restrict__ row,
                            const long long* __restrict__ col,
                            const float* __restrict__ norm,
                            float* __restrict__ dst, int E) {
    const int FQ = 1 << (FSHIFT - 2);              // float4 chunks per edge
    size_t i = (size_t)blockIdx.x * blockDim.x + threadIdx.x;
    size_t e = i >> (FSHIFT - 2);
    if (e >= (size_t)E) return;
    int f4 = (int)(i & (FQ - 1)) << 2;
    long long r = row[e], c = col[e];
    float nv = norm[e];
    v4f v = *(const v4f*)(src + ((size_t)r << FSHIFT) + f4);
    float* d = dst + ((size_t)c << FSHIFT) + f4;
    atomicAdd(d + 0, v.x * nv);
    atomicAdd(d + 1, v.y * nv);
    atomicAdd(d + 2, v.z * nv);
    atomicAdd(d + 3, v.w * nv);
}

// ---------------- WMMA f32 GEMM: C[M,N] = A[M,K] @ B[K,N] (+bias)(+relu) ----------------
// Block: 128 threads = 4 waves. Block tile 64x64. Each wave: one 16-row stripe,
// 4 accumulators covering 64 cols. K staged through LDS in slabs of 32.

#define BM 64
#define BN 64
#define KB 32
#define KBP 36   // padded row stride (floats): 144B = multiple of 16B, spreads banks

template <bool BIAS, bool RELU>
__global__ __launch_bounds__(128) void k_gemm_wmma(
    const float* __restrict__ A, const float* __restrict__ B,
    const float* __restrict__ bias, float* __restrict__ C,
    int M, int K, int N)
{
    __shared__ float As[BM][KBP];   // A tile, row-major [m][k]
    __shared__ float Bs[BN][KBP];   // B tile transposed [n][k]

    const int tid  = threadIdx.x;
    const int wave = tid >> 5;
    const int lane = tid & 31;
    const int m0 = blockIdx.y * BM;
    const int n0 = blockIdx.x * BN;

    const int fm = lane & 15;        // row (A) / col (B) within 16
    const int kb = (lane >> 4) * 2;  // K sub-offset per lane half
    const int wm = wave * 16;        // wave's row stripe within block

    v8f acc[4] = {};

    for (int k0 = 0; k0 < K; k0 += KB) {
        // ---- stage A: BM x KB = 512 float4s, 4 per thread, coalesced along K
        #pragma unroll
        for (int i = 0; i < 4; ++i) {
            int idx = tid + i * 128;        // 0..511
            int r   = idx >> 3;             // KB/4 = 8 float4 per row
            int c4  = (idx & 7) << 2;
            int gr  = m0 + r;
            v4f v = {};
            if (gr < M) v = *(const v4f*)(A + (size_t)gr * K + k0 + c4);
            *(v4f*)(&As[r][c4]) = v;
        }
        // ---- stage B transposed: read float4 along N, scatter into Bs[n][k]
        #pragma unroll
        for (int i = 0; i < 4; ++i) {
            int idx = tid + i * 128;        // 0..511
            int kk  = idx >> 4;             // BN/4 = 16 float4 per k-row
            int c4  = (idx & 15) << 2;
            v4f v = *(const v4f*)(B + (size_t)(k0 + kk) * N + n0 + c4);
            Bs[c4 + 0][kk] = v.x;
            Bs[c4 + 1][kk] = v.y;
            Bs[c4 + 2][kk] = v.z;
            Bs[c4 + 3][kk] = v.w;
        }
        __syncthreads();

        // ---- compute: 8 K-steps of 4, A frag reused across 4 N-tiles
        #pragma unroll
        for (int kk = 0; kk < KB; kk += 4) {
            v2f a = *(const v2f*)(&As[wm + fm][kk + kb]);
            #pragma unroll
            for (int nt = 0; nt < 4; ++nt) {
                v2f b = *(const v2f*)(&Bs[nt * 16 + fm][kk + kb]);
                acc[nt] = __builtin_amdgcn_wmma_f32_16x16x4_f32(
                    /*neg_a=*/false, a, /*neg_b=*/false, b,
                    /*c_mod=*/(short)0, acc[nt],
                    /*reuse_a=*/false, /*reuse_b=*/false);
            }
        }
        __syncthreads();
    }

    // ---- epilogue: C/D layout: vgpr i -> row (i + 8*laneHalf), col = lane&15
    const int half = lane >> 4;
    #pragma unroll
    for (int nt = 0; nt < 4; ++nt) {
        int gc = n0 + nt * 16 + fm;
        float bv = BIAS ? bias[gc] : 0.0f;
        #pragma unroll
        for (int i = 0; i < 8; ++i) {
            int gr = m0 + wm + half * 8 + i;
            if (gr < M) {
                float v = acc[nt][i] + bv;
                if (RELU) v = fmaxf(v, 0.0f);
                C[(size_t)gr * N + gc] = v;
            }
        }
    }
}

// ---------------- host launcher ----------------

extern "C" void kernel_launch(void* const* d_in, const int* in_sizes, int n_in,
                              void* d_out, int out_size, void* d_ws, size_t ws_size,
                              hipStream_t stream) {
    const float*     x  = (const float*)d_in[0];
    const long long* ei = (const long long*)d_in[1];   // int64 per reference
    const float*     W1 = (const float*)d_in[2];
    const float*     b1 = (const float*)d_in[3];
    const float*     W2 = (const float*)d_in[4];
    const float*     b2 = (const float*)d_in[5];
    float* out = (float*)d_out;
    float* ws  = (float*)d_ws;

    const long long* row = ei;
    const long long* col = ei + N_EDGES;

    // workspace layout (float offsets, 256B aligned)
    float* deg  = ws + 0;           //  50000
    float* dinv = ws + 50048;       //  50000
    float* nrm  = ws + 100096;      // 800000
    float* agg1 = ws + 900096;      // 50000*128
    float* h    = ws + 7300096;     // 50000*256
    float* t    = ws + 20100096;    // 50000*64   (end: 23300096 floats ~ 93 MB)

    const int TB = 256;

    // zero deg + agg1 (only buffers that accumulate from cleared state)
    k_zero<<<(50048 + TB - 1) / TB, TB, 0, stream>>>(deg, 50048);
    {
        int n = N_NODES * F_INF;
        k_zero<<<(n + TB - 1) / TB, TB, 0, stream>>>(agg1, n);
    }

    // degree / dinv / edge norms
    k_deg <<<(N_EDGES + TB - 1) / TB, TB, 0, stream>>>(col, deg, N_EDGES);
    k_dinv<<<(N_NODES + TB - 1) / TB, TB, 0, stream>>>(deg, dinv, N_NODES);
    k_norm<<<(N_EDGES + TB - 1) / TB, TB, 0, stream>>>(row, col, dinv, nrm, N_EDGES);

    // layer 1: aggregate first (128 feats < 256) : agg1 = A_norm @ X
    {
        size_t threads = (size_t)N_EDGES * (F_INF / 4);
        k_aggregate<7><<<(unsigned)((threads + TB - 1) / TB), TB, 0, stream>>>(
            x, row, col, nrm, agg1, N_EDGES);
    }

    // h = relu(agg1 @ W1 + b1)   [50000x128 @ 128x256]
    {
        dim3 grid(HIDDENF / BN, (N_NODES + BM - 1) / BM);
        k_gemm_wmma<true, true><<<grid, 128, 0, stream>>>(
            agg1, W1, b1, h, N_NODES, F_INF, HIDDENF);
    }

    // layer 2: GEMM first (64 feats < 256) : t = h @ W2   [50000x256 @ 256x64]
    {
        dim3 grid(F_OUTF / BN, (N_NODES + BM - 1) / BM);
        k_gemm_wmma<false, false><<<grid, 128, 0, stream>>>(
            h, W2, nullptr, t, N_NODES, HIDDENF, F_OUTF);
    }

    // out = b2 broadcast, then out[col] += t[row] * norm
    {
        int n = N_NODES * F_OUTF;
        k_init_out<<<(n + TB - 1) / TB, TB, 0, stream>>>(out, b2, n);
        size_t threads = (size_t)N_EDGES * (F_OUTF / 4);
        k_aggregate<6><<<(unsigned)((threads + TB - 1) / TB), TB, 0, stream>>>(
            t, row, col, nrm, out, N_EDGES);
    }
}